// MultiDense_26130581028993
// MI455X (gfx1250) — compile-verified
//
#include <hip/hip_runtime.h>
#include <hip/hip_bf16.h>

typedef __attribute__((ext_vector_type(2))) float v2f;
typedef __attribute__((ext_vector_type(8))) float v8f;

#define NDIMS 64
#define DIN   64
#define DOUT  64
#define TILE_M 16

// Workspace layout (as int):
//   [0,   64)  per-expert counts
//   [64, 128)  per-expert fill cursors (atomics)
//   [128,193)  padded segment offsets (65 entries, 16-row aligned)
//   [256, ...) perm[] : sample indices grouped by expert

__global__ void md_init(int* __restrict__ ws) {
    int i = threadIdx.x;
    if (i < NDIMS) { ws[i] = 0; ws[64 + i] = 0; }
}

__global__ void md_hist(const long long* __restrict__ lookups,
                        int* __restrict__ counts, int n) {
    int i = blockIdx.x * blockDim.x + threadIdx.x;
    if (i < n) atomicAdd(&counts[(int)lookups[i]], 1);
}

__global__ void md_scan(int* __restrict__ ws) {
    if (threadIdx.x == 0) {
        int acc = 0;
        for (int e = 0; e < NDIMS; ++e) {
            ws[128 + e] = acc;
            acc += (ws[e] + TILE_M - 1) & ~(TILE_M - 1);  // 16-row aligned segments
        }
        ws[128 + NDIMS] = acc;
    }
}

__global__ void md_scatter(const long long* __restrict__ lookups,
                           int* __restrict__ ws, int n) {
    int i = blockIdx.x * blockDim.x + threadIdx.x;
    if (i < n) {
        int e = (int)lookups[i];
        int p = atomicAdd(&ws[64 + e], 1);
        ws[256 + ws[128 + e] + p] = i;
    }
}

// One wave32 per 16-sample tile; computes 16x64 outputs via V_WMMA_F32_16X16X4_F32.
__global__ __launch_bounds__(32) void md_gemm(
        const float* __restrict__ values,
        const float* __restrict__ weights,
        const float* __restrict__ bias,
        const int*   __restrict__ ws,
        float*       __restrict__ out) {
    __shared__ int   s_row[TILE_M];
    __shared__ float s_a[TILE_M * 68];   // stride 68 floats: 16B aligned, conflict-free

    const int lane = threadIdx.x;
    const int R    = blockIdx.x * TILE_M;
    const int* poff = ws + 128;
    if (R >= poff[NDIMS]) return;        // wave-uniform exit

    // Locate owning expert (uniform scalar loop over 64 offsets)
    int e = 0;
    while (e < NDIMS - 1 && R >= poff[e + 1]) ++e;
    const int  cnt  = ws[e];
    const int* perm = ws + 256;

    if (lane < TILE_M) {
        int local = R - poff[e] + lane;
        s_row[lane] = (local < cnt) ? perm[R + lane] : -1;   // -1 => padding row
    }
    __syncthreads();

    const float* W = weights + (size_t)e * (DIN * DOUT);
    __builtin_prefetch(W, 0, 3);   // global_prefetch_b8: warm expert weights

    // Stage A tile (16 x 64 f32) into LDS with coalesced float4 loads
    for (int t = lane; t < TILE_M * 16; t += 32) {
        int row = t >> 4, c4 = t & 15;
        int src = s_row[row];
        float4 v = make_float4(0.f, 0.f, 0.f, 0.f);
        if (src >= 0) v = *(const float4*)(values + (size_t)src * DIN + c4 * 4);
        *(float4*)(&s_a[row * 68 + c4 * 4]) = v;
    }
    __syncthreads();

    const int hlf  = lane >> 4;      // 0: lanes 0-15, 1: lanes 16-31
    const int col  = lane & 15;      // N within 16-wide tile / M for A fragment
    const int koff = hlf * 2;        // K sub-offset per ISA A/B fragment layout

    v8f acc[4];
#pragma unroll
    for (int nt = 0; nt < 4; ++nt) { // C init = bias (depends only on N column)
        float b = bias[e * DOUT + nt * 16 + col];
        acc[nt] = (v8f){b, b, b, b, b, b, b, b};
    }

#pragma unroll
    for (int kk = 0; kk < DIN / 4; ++kk) {
        // A frag: lane holds A[M=col][4kk+koff .. +1] (contiguous -> ds b64)
        v2f a = *(const v2f*)(&s_a[col * 68 + kk * 4 + koff]);
        const float* wk = W + (kk * 4 + koff) * DOUT;
#pragma unroll
        for (int nt = 0; nt < 4; ++nt) {
            // B frag: VGPR0 = B[K=4kk+koff][N], VGPR1 = B[K=4kk+koff+1][N]
            v2f b;
            b.x = wk[nt * 16 + col];
            b.y = wk[DOUT + nt * 16 + col];
            acc[nt] = __builtin_amdgcn_wmma_f32_16x16x4_f32(
                false, a, false, b, (short)0, acc[nt], false, false);
        }
    }

    // D layout: VGPR r holds M=r (lanes 0-15) / M=r+8 (lanes 16-31), N=col
#pragma unroll
    for (int nt = 0; nt < 4; ++nt) {
#pragma unroll
        for (int r = 0; r < 8; ++r) {
            int m    = r + hlf * 8;
            int srow = s_row[m];
            if (srow >= 0) {
                float v = acc[nt][r];
                out[(size_t)srow * DOUT + nt * 16 + col] = v > 0.f ? v : 0.f;
            }
        }
    }
}

extern "C" void kernel_launch(void* const* d_in, const int* in_sizes, int n_in,
                              void* d_out, int out_size, void* d_ws, size_t ws_size,
                              hipStream_t stream) {
    const long long* lookups = (const long long*)d_in[0];  // int64
    const float*     values  = (const float*)d_in[1];
    const float*     weights = (const float*)d_in[2];
    const float*     bias    = (const float*)d_in[3];
    float* out = (float*)d_out;
    int*   ws  = (int*)d_ws;
    const int n = in_sizes[0];

    md_init<<<1, 64, 0, stream>>>(ws);
    md_hist<<<(n + 255) / 256, 256, 0, stream>>>(lookups, ws, n);
    md_scan<<<1, 1, 0, stream>>>(ws);
    md_scatter<<<(n + 255) / 256, 256, 0, stream>>>(lookups, ws, n);

    // Enough tiles to cover all padded segments (padding <= 15 rows per expert)
    const int tiles = (n + TILE_M - 1) / TILE_M + NDIMS;
    md_gemm<<<tiles, 32, 0, stream>>>(values, weights, bias, ws, out);
}